// VoxelEncoding_16329465659649
// MI455X (gfx1250) — compile-verified
//
#include <hip/hip_runtime.h>

// VoxelEncoding: trilinear interpolation of gathered corner embeddings.
// out[n, d] = sum_{c=0..7} w[n,c] * embeddings[center2corner[p2v_idx[n]][c], d]
// w[n,c]    = prod_dim (bit ? p : 1-p),  p = (pts[n]-cp[v])/voxel_size + 0.5
// corner c -> (x,y,z) bits = ((c>>2)&1, (c>>1)&1, c&1)   [meshgrid 'ij' order]
//
// Memory-bound gather kernel: tables (6+16+64 MB) are L2-resident (192 MB L2),
// streaming traffic ~160 MB -> ~7us floor at 23.3 TB/s. WMMA inapplicable:
// the 8x16 "B matrix" is gathered per point, never wave-shared.

#define EMBED_DIM 16

typedef float float4v __attribute__((ext_vector_type(4)));

// Compute one point's 4-channel output quad (channels lane4*4 .. lane4*4+3).
__device__ __forceinline__ float4v interp_point(
    int n, float inv_vs,
    const float* __restrict__ pts,   // (N,3)
    const int*   __restrict__ p2v,   // (N,)
    const float* __restrict__ embL,  // emb + lane4*4 (pre-offset base)
    const float* __restrict__ cpt,   // (V,3)
    const int*   __restrict__ c2c)   // (V,8)
{
    const int v = p2v[n];

    const float px = (pts[(size_t)n * 3 + 0] - cpt[(size_t)v * 3 + 0]) * inv_vs + 0.5f;
    const float py = (pts[(size_t)n * 3 + 1] - cpt[(size_t)v * 3 + 1]) * inv_vs + 0.5f;
    const float pz = (pts[(size_t)n * 3 + 2] - cpt[(size_t)v * 3 + 2]) * inv_vs + 0.5f;

    // corner index row: 32B-aligned -> two b128 loads
    const int4* crow = (const int4*)(c2c + (size_t)v * 8);
    const int4 ca = crow[0];
    const int4 cb = crow[1];
    const int idx[8] = { ca.x, ca.y, ca.z, ca.w, cb.x, cb.y, cb.z, cb.w };

    const float x0 = 1.0f - px, x1 = px;
    const float y0 = 1.0f - py, y1 = py;
    const float z0 = 1.0f - pz, z1 = pz;
    const float w[8] = { x0*y0*z0, x0*y0*z1, x0*y1*z0, x0*y1*z1,
                         x1*y0*z0, x1*y0*z1, x1*y1*z0, x1*y1*z1 };

    // 8 independent b128 gathers from the L2-resident embedding table
    float4v acc0 = {0.f, 0.f, 0.f, 0.f};
    float4v acc1 = {0.f, 0.f, 0.f, 0.f};
    #pragma unroll
    for (int c = 0; c < 8; c += 2) {
        const float4v e0 = *(const float4v*)(embL + (size_t)idx[c]     * EMBED_DIM);
        const float4v e1 = *(const float4v*)(embL + (size_t)idx[c + 1] * EMBED_DIM);
        acc0 += w[c]     * e0;
        acc1 += w[c + 1] * e1;
    }
    return acc0 + acc1;
}

// Uniform hot path: n_pts divisible by PPT, grid sized so every group handles
// exactly PPT points. Fully unrolled -> ~8*PPT gathers in flight per thread.
template <int PPT>
__global__ __launch_bounds__(320) void voxel_interp_uniform(
    const float* __restrict__ pts, const int* __restrict__ p2v,
    const float* __restrict__ emb, const float* __restrict__ cpt,
    const int*   __restrict__ c2c, const float* __restrict__ vsz,
    float*       __restrict__ out, int groups) // groups == n_pts / PPT
{
    const int tid   = blockIdx.x * blockDim.x + threadIdx.x;
    const int lane4 = tid & 3;
    const int g0    = tid >> 2;

    const float inv_vs = 1.0f / vsz[0];
    const float* __restrict__ embL = emb + lane4 * 4;
    float*       __restrict__ outL = out + lane4 * 4;

    #pragma unroll
    for (int i = 0; i < PPT; ++i) {
        const int n = g0 + i * groups;
        const float4v r = interp_point(n, inv_vs, pts, p2v, embL, cpt, c2c);
        // coalesced: a wave's 32 lanes cover 8 points x 64B contiguous
        *(float4v*)(outL + (size_t)n * EMBED_DIM) = r;
    }
}

// Generic fallback: grid-stride with software-pipelined prefetch of the next
// iteration's streaming inputs (gfx1250 global_prefetch_b8 path).
__global__ __launch_bounds__(256) void voxel_interp_generic(
    const float* __restrict__ pts, const int* __restrict__ p2v,
    const float* __restrict__ emb, const float* __restrict__ cpt,
    const int*   __restrict__ c2c, const float* __restrict__ vsz,
    float*       __restrict__ out, int n_pts, int groups_total)
{
    const int tid   = blockIdx.x * blockDim.x + threadIdx.x;
    const int lane4 = tid & 3;
    const int g0    = tid >> 2;

    const float inv_vs = 1.0f / vsz[0];
    const float* __restrict__ embL = emb + lane4 * 4;
    float*       __restrict__ outL = out + lane4 * 4;

    for (int n = g0; n < n_pts; n += groups_total) {
        const int n_next = n + groups_total;
        if (n_next < n_pts) {
            __builtin_prefetch(&pts[(size_t)n_next * 3], 0, 3);
            __builtin_prefetch(&p2v[n_next], 0, 3);
        }
        const float4v r = interp_point(n, inv_vs, pts, p2v, embL, cpt, c2c);
        *(float4v*)(outL + (size_t)n * EMBED_DIM) = r;
    }
}

extern "C" void kernel_launch(void* const* d_in, const int* in_sizes, int n_in,
                              void* d_out, int out_size, void* d_ws, size_t ws_size,
                              hipStream_t stream) {
    const float* pts = (const float*)d_in[0];  // (N_PTS, 3) f32
    const int*   p2v = (const int*)  d_in[1];  // (N_PTS,)   i32
    const float* emb = (const float*)d_in[2];  // (N_EMB,16) f32
    const float* cpt = (const float*)d_in[3];  // (N_VOX, 3) f32
    const int*   c2c = (const int*)  d_in[4];  // (N_VOX, 8) i32
    const float* vsz = (const float*)d_in[5];  // (1,)       f32
    float*       out = (float*)d_out;          // (N_PTS,16) f32

    const int n_pts = in_sizes[1];             // p2v_idx element count == N_PTS

    constexpr int PPT   = 4;                   // points per thread (hot path)
    constexpr int BLOCK = 320;                 // 10 waves, 80 point-groups/block
    constexpr int GPB   = BLOCK / 4;

    if (n_pts % PPT == 0 && (n_pts / PPT) % GPB == 0) {
        // e.g. N=2,000,000: groups=500,000 -> grid=6250, every group does 4 pts
        const int groups = n_pts / PPT;
        const int grid   = groups / GPB;
        voxel_interp_uniform<PPT><<<grid, BLOCK, 0, stream>>>(
            pts, p2v, emb, cpt, c2c, vsz, out, groups);
    } else {
        const int block = 256;
        const long long groups_needed = ((long long)n_pts + PPT - 1) / PPT;
        const long long threads = groups_needed * 4;
        const int grid = (int)((threads + block - 1) / block);
        const int groups_total = grid * (block / 4);
        voxel_interp_generic<<<grid, block, 0, stream>>>(
            pts, p2v, emb, cpt, c2c, vsz, out, n_pts, groups_total);
    }
}